// MergeDNA_73177652789841
// MI455X (gfx1250) — compile-verified
//
#include <hip/hip_runtime.h>
#include <math.h>

// Problem constants (match reference)
#define T_TOK   8192
#define D_DIM   256
#define N_DIM   8192
#define N_DIM4  (N_DIM / 4)            // 2048 float4 per s-row
#define WINDOW  512
#define NWIN    (T_TOK / WINDOW)       // 16
#define A_HALF  (WINDOW / 2)           // 256
#define NSEL    128                    // min(512*0.25, 256)
#define NUNM    (A_HALF - NSEL)        // 128
#define ROWS_PER_WIN (WINDOW - NSEL)   // 384
#define X_ROWS  (NWIN * ROWS_PER_WIN)  // 6144
#define EPSF    1e-6f
#define NEG_INF (-3.402823466e38f)

typedef __attribute__((ext_vector_type(2))) float v2f;
typedef __attribute__((ext_vector_type(8))) float v8f;

// ---------------------------------------------------------------------------
// K1: Kmat = x @ Wk  (fp32 WMMA, one wave per 16x16 output tile)
// A 16x4 f32 layout: lanes 0-15 row M=0..15 hold K={0,1}; lanes 16-31 hold K={2,3}
// B 4x16 f32 layout (mirror): lanes 0-15 N=0..15 hold K={0,1}; lanes 16-31 K={2,3}
// C/D: VGPR i, lanes 0-15 -> (M=i,   N=lane); lanes 16-31 -> (M=i+8, N=lane-16)
// ---------------------------------------------------------------------------
__global__ __launch_bounds__(256) void k_gemm(const float* __restrict__ x,
                                              const float* __restrict__ Wk,
                                              float* __restrict__ Kmat) {
    const int wave = threadIdx.x >> 5;
    const int lane = threadIdx.x & 31;
    const int half = lane >> 4, l16 = lane & 15;
    const int tile  = blockIdx.x * 8 + wave;     // 512 mtiles * 16 ntiles = 8192
    const int mtile = tile >> 4;
    const int ntile = tile & 15;
    const int row = mtile * 16 + l16;            // A-fragment row for this lane
    const int col = ntile * 16 + l16;            // B-fragment column for this lane

    v8f acc = {};
    for (int k = 0; k < D_DIM; k += 4) {
        v2f a, b;
        const float2 av = *reinterpret_cast<const float2*>(x + row * D_DIM + k + 2 * half);
        a.x = av.x; a.y = av.y;
        b.x = Wk[(k + 2 * half + 0) * D_DIM + col];
        b.y = Wk[(k + 2 * half + 1) * D_DIM + col];
        acc = __builtin_amdgcn_wmma_f32_16x16x4_f32(false, a, false, b,
                                                    (short)0, acc, false, false);
    }
#pragma unroll
    for (int i = 0; i < 8; ++i)
        Kmat[(mtile * 16 + i + 8 * half) * D_DIM + ntile * 16 + l16] = acc[i];
}

// ---------------------------------------------------------------------------
// K2: row-normalize Kmat (wave per row)
// ---------------------------------------------------------------------------
__global__ __launch_bounds__(256) void k_norm(float* __restrict__ Kmat) {
    const int wave = threadIdx.x >> 5, lane = threadIdx.x & 31;
    const int row = blockIdx.x * 8 + wave;
    float v[8];
    float ss = 0.f;
#pragma unroll
    for (int i = 0; i < 8; ++i) {
        v[i] = Kmat[row * D_DIM + lane + 32 * i];
        ss += v[i] * v[i];
    }
#pragma unroll
    for (int m = 16; m >= 1; m >>= 1) ss += __shfl_xor(ss, m, 32);
    const float scale = 1.f / fmaxf(sqrtf(ss), EPSF);
#pragma unroll
    for (int i = 0; i < 8; ++i)
        Kmat[row * D_DIM + lane + 32 * i] = v[i] * scale;
}

// ---------------------------------------------------------------------------
// K3: rowsum[r] = sum(s[r, :])  (wave per row, float4 / b128 coalesced stream)
// ---------------------------------------------------------------------------
__global__ __launch_bounds__(256) void k_rowsum(const float* __restrict__ s,
                                                float* __restrict__ rowsum) {
    const int wave = threadIdx.x >> 5, lane = threadIdx.x & 31;
    const int row = blockIdx.x * 8 + wave;
    const float4* __restrict__ srow4 =
        reinterpret_cast<const float4*>(s + (size_t)row * N_DIM);
    float acc = 0.f;
    for (int c = lane; c < N_DIM4; c += 32) {
        const float4 v = srow4[c];
        acc += (v.x + v.y) + (v.z + v.w);
    }
#pragma unroll
    for (int m = 16; m >= 1; m >>= 1) acc += __shfl_xor(acc, m, 32);
    if (lane == 0) rowsum[row] = acc;
}

// ---------------------------------------------------------------------------
// K4: per-window scores = ka @ kb^T via WMMA f32, fused row max/argmax.
// One wave per (window, 16-row strip): 16 win * 16 strips = 256 wave tasks.
// ---------------------------------------------------------------------------
__global__ __launch_bounds__(256) void k_scores(const float* __restrict__ Kmat,
                                                float* __restrict__ bestS,
                                                int* __restrict__ bestD) {
    const int wave = threadIdx.x >> 5;
    const int lane = threadIdx.x & 31;
    const int half = lane >> 4, l16 = lane & 15;
    const int task  = blockIdx.x * 8 + wave;     // 0..255
    const int win   = task >> 4;
    const int strip = task & 15;

    const int m  = strip * 16 + l16;             // a-row this lane loads for A
    const int ga = win * WINDOW + 2 * m;         // global token of a-row

    float bestv[8];
    int   besti[8];
#pragma unroll
    for (int i = 0; i < 8; ++i) { bestv[i] = NEG_INF; besti[i] = 0; }

    for (int ntile = 0; ntile < 16; ++ntile) {
        const int n  = ntile * 16 + l16;         // b-column this lane loads for B
        const int gb = win * WINDOW + 2 * n + 1; // global token of b-row (B = kb^T)
        v8f acc = {};
        for (int k = 0; k < D_DIM; k += 4) {
            v2f a, b;
            const float2 av = *reinterpret_cast<const float2*>(Kmat + ga * D_DIM + k + 2 * half);
            const float2 bv = *reinterpret_cast<const float2*>(Kmat + gb * D_DIM + k + 2 * half);
            a.x = av.x; a.y = av.y;
            b.x = bv.x; b.y = bv.y;
            acc = __builtin_amdgcn_wmma_f32_16x16x4_f32(false, a, false, b,
                                                        (short)0, acc, false, false);
        }
        // C layout: lane holds column N = l16 (+16*ntile globally) for 8 rows
        const int nidx = ntile * 16 + l16;
#pragma unroll
        for (int i = 0; i < 8; ++i) {
            const float v = acc[i];
            if (v > bestv[i]) { bestv[i] = v; besti[i] = nidx; } // ties keep lower idx
        }
    }

    // Cross-lane argmax within each 16-lane half (rows i+8*half); tie -> lower idx
#pragma unroll
    for (int i = 0; i < 8; ++i) {
        float v  = bestv[i];
        int   ix = besti[i];
#pragma unroll
        for (int msk = 8; msk >= 1; msk >>= 1) {
            const float ov = __shfl_xor(v, msk, 32);
            const int   oi = __shfl_xor(ix, msk, 32);
            if (ov > v || (ov == v && oi < ix)) { v = ov; ix = oi; }
        }
        if (l16 == 0) {
            const int rowm = strip * 16 + i + 8 * half;
            bestS[win * A_HALF + rowm] = v;
            bestD[win * A_HALF + rowm] = ix;
        }
    }
}

// ---------------------------------------------------------------------------
// K5: top-NSEL selection per window (stable, ties -> lower index) + compaction
// ---------------------------------------------------------------------------
__global__ __launch_bounds__(256) void k_select(const float* __restrict__ bestS,
                                                const int* __restrict__ bestD,
                                                int* __restrict__ srcIdx,
                                                int* __restrict__ srcDst,
                                                int* __restrict__ unmIdx) {
    __shared__ float Ssc[A_HALF];
    __shared__ int   Sdst[A_HALF];
    __shared__ int   Ssel[A_HALF];
    const int win = blockIdx.x, t = threadIdx.x;
    Ssc[t]  = bestS[win * A_HALF + t];
    Sdst[t] = bestD[win * A_HALF + t];
    __syncthreads();

    const float mys = Ssc[t];
    int rank = 0;
    for (int j = 0; j < A_HALF; ++j) {
        const float o = Ssc[j];
        rank += (o > mys) || (o == mys && j < t);
    }
    const int sel = (rank < NSEL) ? 1 : 0;
    Ssel[t] = sel;
    __syncthreads();

    int pos = 0;
    for (int j = 0; j < t; ++j) pos += (Ssel[j] == sel);
    if (sel) {
        srcIdx[win * NSEL + pos] = t;
        srcDst[win * NSEL + pos] = Sdst[t];
    } else {
        unmIdx[win * NUNM + pos] = t;   // ascending index order preserved
    }
}

// ---------------------------------------------------------------------------
// K6: new_x rows (6144 blocks, one output row each; 256 threads = D columns)
// ---------------------------------------------------------------------------
__global__ __launch_bounds__(256) void k_xout(const float* __restrict__ x,
                                              const float* __restrict__ rowsum,
                                              const int* __restrict__ srcIdx,
                                              const int* __restrict__ srcDst,
                                              const int* __restrict__ unmIdx,
                                              float* __restrict__ outx) {
    __shared__ int sIdx[NSEL], sDst[NSEL];
    const int r   = blockIdx.x;
    const int win = r / ROWS_PER_WIN, rl = r % ROWS_PER_WIN;
    const int t   = threadIdx.x;
    if (t < NSEL) {
        sIdx[t] = srcIdx[win * NSEL + t];
        sDst[t] = srcDst[win * NSEL + t];
    }
    __syncthreads();

    const int base = win * WINDOW;
    float out;
    if (rl < NUNM) {                               // unmerged a-token: plain copy
        const int g = base + 2 * unmIdx[win * NUNM + rl];
        out = x[g * D_DIM + t];
    } else {                                       // merged b-token
        const int j  = rl - NUNM;
        const int gb = base + 2 * j + 1;
        const float bsz = rowsum[gb];
        float acc = x[gb * D_DIM + t] * bsz;
        float szn = bsz;
        for (int m = 0; m < NSEL; ++m) {           // block-uniform branch
            if (sDst[m] == j) {
                const int gs = base + 2 * sIdx[m];
                const float asz = rowsum[gs];
                acc += x[gs * D_DIM + t] * asz;
                szn += asz;
            }
        }
        out = acc / fmaxf(szn, EPSF);
    }
    outx[(size_t)r * D_DIM + t] = out;
}

// ---------------------------------------------------------------------------
// K7: new_s rows (6144 blocks; 8192-wide rows streamed as float4 / b128,
//     8 vec4 per thread, fully coalesced)
// ---------------------------------------------------------------------------
__global__ __launch_bounds__(256) void k_sout(const float* __restrict__ s,
                                              const int* __restrict__ srcIdx,
                                              const int* __restrict__ srcDst,
                                              const int* __restrict__ unmIdx,
                                              float* __restrict__ outs) {
    __shared__ int sIdx[NSEL], sDst[NSEL];
    __shared__ int matches[NSEL];
    __shared__ int nMatch;
    const int r   = blockIdx.x;
    const int win = r / ROWS_PER_WIN, rl = r % ROWS_PER_WIN;
    const int t   = threadIdx.x;
    if (t < NSEL) {
        sIdx[t] = srcIdx[win * NSEL + t];
        sDst[t] = srcDst[win * NSEL + t];
    }
    __syncthreads();

    const int base = win * WINDOW;
    const float4* __restrict__ s4 = reinterpret_cast<const float4*>(s);
    float4* __restrict__ orow4 =
        reinterpret_cast<float4*>(outs + (size_t)r * N_DIM);

    if (rl < NUNM) {                               // unmerged a-token: copy s row
        const int g = base + 2 * unmIdx[win * NUNM + rl];
        const float4* __restrict__ srow4 = s4 + (size_t)g * N_DIM4;
        for (int c = t; c < N_DIM4; c += 256) orow4[c] = srow4[c];
    } else {                                       // merged b-token: gather-add
        const int j  = rl - NUNM;
        const int gb = base + 2 * j + 1;
        if (t == 0) {                              // compact matching src tokens
            int n = 0;
            for (int m = 0; m < NSEL; ++m)
                if (sDst[m] == j) matches[n++] = base + 2 * sIdx[m];
            nMatch = n;
        }
        __syncthreads();
        const int nm = nMatch;
        const float4* __restrict__ brow4 = s4 + (size_t)gb * N_DIM4;
        for (int c = t; c < N_DIM4; c += 256) {
            float4 acc = brow4[c];
            for (int m = 0; m < nm; ++m) {
                const float4 v = s4[(size_t)matches[m] * N_DIM4 + c];
                acc.x += v.x; acc.y += v.y; acc.z += v.z; acc.w += v.w;
            }
            orow4[c] = acc;
        }
    }
}

// ---------------------------------------------------------------------------
extern "C" void kernel_launch(void* const* d_in, const int* in_sizes, int n_in,
                              void* d_out, int out_size, void* d_ws, size_t ws_size,
                              hipStream_t stream) {
    const float* x  = (const float*)d_in[0];   // [8192, 256]
    const float* s  = (const float*)d_in[1];   // [8192, 8192]
    const float* Wk = (const float*)d_in[2];   // [256, 256]
    float* out = (float*)d_out;                // new_x (6144*256) ++ new_s (6144*8192)

    // Workspace carve-up (~8.5 MB, everything written before read)
    float* Kmat   = (float*)d_ws;                    // 8192*256 f32
    float* rowsum = Kmat + (size_t)T_TOK * D_DIM;    // 8192 f32
    float* bestS  = rowsum + T_TOK;                  // 16*256 f32
    int*   bestD  = (int*)(bestS + NWIN * A_HALF);   // 16*256 i32
    int*   srcIdx = bestD + NWIN * A_HALF;           // 16*128 i32
    int*   srcDst = srcIdx + NWIN * NSEL;            // 16*128 i32
    int*   unmIdx = srcDst + NWIN * NSEL;            // 16*128 i32

    k_gemm  <<<1024, 256, 0, stream>>>(x, Wk, Kmat);
    k_norm  <<<1024, 256, 0, stream>>>(Kmat);
    k_rowsum<<<1024, 256, 0, stream>>>(s, rowsum);
    k_scores<<<  32, 256, 0, stream>>>(Kmat, bestS, bestD);
    k_select<<<NWIN, 256, 0, stream>>>(bestS, bestD, srcIdx, srcDst, unmIdx);
    k_xout  <<<X_ROWS, 256, 0, stream>>>(x, rowsum, srcIdx, srcDst, unmIdx, out);
    k_sout  <<<X_ROWS, 256, 0, stream>>>(s, srcIdx, srcDst, unmIdx,
                                         out + (size_t)X_ROWS * D_DIM);
}